// GraphClassifier_44624710205918
// MI455X (gfx1250) — compile-verified
//
#include <hip/hip_runtime.h>
#include <hip/hip_bf16.h>

typedef float v2f __attribute__((ext_vector_type(2)));
typedef float v8f __attribute__((ext_vector_type(8)));

#define WMMA_F32(a, b, c) \
  __builtin_amdgcn_wmma_f32_16x16x4_f32(false, (a), false, (b), (short)0, (c), false, false)

// ---------------------------------------------------------------------------
// Constants (match reference)
// ---------------------------------------------------------------------------
constexpr int B_   = 2048;
constexpr int N0_  = 128;
constexpr int CIN_ = 128;
constexpr int DIM_ = 64;
constexpr int EPG_ = N0_ * 16;        // 2048 edges per graph, contiguous
constexpr long E_  = (long)B_ * EPG_; // 4,194,304

// ---------------------------------------------------------------------------
// Kernel 1: C[M,64] = A[M,K] @ B[K,64] with fp32 WMMA (16x16x4).
// One block = 64 rows, 128 threads (4 waves), each wave: 16 rows x 64 cols.
// A/B staged in LDS with padded strides for conflict-free fragment reads.
// A fragment (16x4 f32): lanes 0-15 hold K=k..k+1, lanes 16-31 K=k+2..k+3.
// C/D layout: VGPR r -> M = r (lanes 0-15) / 8+r (lanes 16-31), N = lane&15.
// ---------------------------------------------------------------------------
template <int K>
__global__ void gemm_wmma_n64(const float* __restrict__ A,
                              const float* __restrict__ Bm,
                              float* __restrict__ C) {
  constexpr int SA = K + 2;   // padded A row stride (kills bank conflicts)
  constexpr int SB = 65;      // padded B row stride
  extern __shared__ float sm[];
  float* sA = sm;             // 64 * SA
  float* sB = sm + 64 * SA;   // K  * SB

  const int tid = threadIdx.x;
  const size_t rbase = (size_t)blockIdx.x * 64;

  // Stage A tile (64 consecutive rows -> contiguous 64*K floats): coalesced.
  for (int i = tid * 4; i < 64 * K; i += 128 * 4) {
    int r = i / K, c = i % K;
    float4 v = *(const float4*)(A + rbase * K + i);
    float* d = sA + r * SA + c;
    d[0] = v.x; d[1] = v.y; d[2] = v.z; d[3] = v.w;
  }
  // Stage all of B (K x 64).
  for (int i = tid * 4; i < K * 64; i += 128 * 4) {
    int r = i >> 6, c = i & 63;
    float4 v = *(const float4*)(Bm + i);
    float* d = sB + r * SB + c;
    d[0] = v.x; d[1] = v.y; d[2] = v.z; d[3] = v.w;
  }
  __syncthreads();

  const int wave = tid >> 5, lane = tid & 31;
  const int half = lane >> 4, l16 = lane & 15;
  const int mb = wave * 16;

  v8f acc0 = {}, acc1 = {}, acc2 = {}, acc3 = {};
  for (int k = 0; k < K; k += 4) {
    const int kk = k + 2 * half;
    v2f a = { sA[(mb + l16) * SA + kk], sA[(mb + l16) * SA + kk + 1] };
    v2f b0 = { sB[kk * SB +  0 + l16], sB[(kk + 1) * SB +  0 + l16] };
    v2f b1 = { sB[kk * SB + 16 + l16], sB[(kk + 1) * SB + 16 + l16] };
    v2f b2 = { sB[kk * SB + 32 + l16], sB[(kk + 1) * SB + 32 + l16] };
    v2f b3 = { sB[kk * SB + 48 + l16], sB[(kk + 1) * SB + 48 + l16] };
    acc0 = WMMA_F32(a, b0, acc0);
    acc1 = WMMA_F32(a, b1, acc1);
    acc2 = WMMA_F32(a, b2, acc2);
    acc3 = WMMA_F32(a, b3, acc3);
  }

  const int mrow = mb + 8 * half;
  float* Cg = C + rbase * 64;
#pragma unroll
  for (int r = 0; r < 8; ++r) {
    Cg[(mrow + r) * 64 +  0 + l16] = acc0[r];
    Cg[(mrow + r) * 64 + 16 + l16] = acc1[r];
    Cg[(mrow + r) * 64 + 32 + l16] = acc2[r];
    Cg[(mrow + r) * 64 + 48 + l16] = acc3[r];
  }
}

// ---------------------------------------------------------------------------
// Kernel 2: per-graph conv1 aggregation + ReLU + TopK pool 1 + readout 1.
// One block (256 threads) per graph; graph's h0 (32KB) and accumulator
// (32KB) live in LDS; edge segment-sum uses ds_add_f32 (bank-rotated).
// ---------------------------------------------------------------------------
__global__ void graph_stage1(const float* __restrict__ h0,
                             const int* __restrict__ esrc,
                             const int* __restrict__ edst,
                             const float* __restrict__ b1,
                             const float* __restrict__ p1,
                             float* __restrict__ xp1,   // [B*64, 64] rank order
                             int* __restrict__ nid_g,   // [B*128] new id or -1
                             float* __restrict__ x1,    // [B, 128] readout
                             float* __restrict__ a1) {  // [B, 64] sigmoid(sc)
  extern __shared__ float sm[];
  float* sh_h   = sm;                 // 128*64 : h0 tile, later gated (rank)
  float* sh_agg = sm + 128 * 64;      // 128*64 : aggregation -> relu(h1)
  float* sh_sc  = sm + 2 * 128 * 64;  // 128 scores
  int*   sh_ni  = (int*)(sh_sc + 128);

  const int g = blockIdx.x, tid = threadIdx.x;

  const float4* h4 = (const float4*)(h0 + (size_t)g * 128 * 64);
  float4* s4 = (float4*)sh_h;
  float4* z4 = (float4*)sh_agg;
  float4 zz = {0.f, 0.f, 0.f, 0.f};
  for (int i = tid; i < 128 * 64 / 4; i += 256) { s4[i] = h4[i]; z4[i] = zz; }
  __syncthreads();

  // segment-sum over this graph's 2048 edges, entirely in LDS
  const long ebase = (long)g * EPG_;
  for (int e = tid; e < EPG_; e += 256) {
    int sl = esrc[ebase + e] - g * N0_;
    int dl = edst[ebase + e] - g * N0_;
    const float* srcv = sh_h + sl * 64;
    float* dstv = sh_agg + dl * 64;
    for (int dd = 0; dd < 64; ++dd) {
      int d = (dd + tid) & 63;  // rotate to spread LDS banks
      atomicAdd(&dstv[d], srcv[d]);
    }
  }
  __syncthreads();

  // h1 = relu(agg + b1) in place
  for (int i = tid; i < 128 * 64; i += 256) {
    float v = sh_agg[i] + b1[i & 63];
    sh_agg[i] = v > 0.f ? v : 0.f;
  }
  __syncthreads();

  // score = tanh((h1 . p1) / ||p1||)
  if (tid < 128) {
    float dp = 0.f, pp = 0.f;
    for (int d = 0; d < 64; ++d) {
      float pv = p1[d];
      dp += sh_agg[tid * 64 + d] * pv;
      pp += pv * pv;
    }
    sh_sc[tid] = tanhf(dp * rsqrtf(pp));
  }
  __syncthreads();

  // stable rank (matches top_k: score desc, index asc on ties); keep top 64
  if (tid < 128) {
    float si = sh_sc[tid];
    int r = 0;
    for (int j = 0; j < 128; ++j) {
      float sj = sh_sc[j];
      r += (sj > si) || (sj == si && j < tid);
    }
    int id = (r < 64) ? r : -1;
    sh_ni[tid] = id;
    nid_g[g * 128 + tid] = id;
    if (id >= 0) a1[(size_t)g * 64 + r] = 1.f / (1.f + expf(-si));
  }
  __syncthreads();

  // gate kept rows by score; emit xp1 in rank order; stash for readout
  for (int i = tid; i < 128 * 64; i += 256) {
    int node = i >> 6, d = i & 63;
    int r = sh_ni[node];
    if (r >= 0) {
      float v = sh_agg[i] * sh_sc[node];
      xp1[((size_t)g * 64 + r) * 64 + d] = v;
      sh_h[r * 64 + d] = v;  // sh_h is free after the edge pass
    }
  }
  __syncthreads();

  // readout: max / mean over 64 kept rows
  if (tid < 64) {
    float mx = -3.402823466e38f, smv = 0.f;
    for (int r = 0; r < 64; ++r) {
      float v = sh_h[r * 64 + tid];
      mx = v > mx ? v : mx;
      smv += v;
    }
    x1[(size_t)g * 128 + tid] = mx;
    x1[(size_t)g * 128 + 64 + tid] = smv * (1.f / 64.f);
  }
}

// ---------------------------------------------------------------------------
// Kernel 3: per-graph conv2 (WMMA m2 = xp1@W2 in-workgroup) + masked edge
// aggregation + ReLU + TopK pool 2 + readout 2 + xg = x1 + x2.
// ---------------------------------------------------------------------------
__global__ void graph_stage2(const float* __restrict__ xp1,
                             const float* __restrict__ W2,
                             const int* __restrict__ esrc,
                             const int* __restrict__ edst,
                             const int* __restrict__ nid_g,
                             const float* __restrict__ b2,
                             const float* __restrict__ p2,
                             const float* __restrict__ x1,
                             float* __restrict__ xg,    // [B, 128]
                             float* __restrict__ a2) {  // [B, 32]
  constexpr int XS = 66, WS = 65;  // padded strides for WMMA frag reads
  extern __shared__ float sm[];
  float* sh_x  = sm;                  // 64*XS : xp1 tile
  float* sh_w  = sh_x + 64 * XS;      // 64*WS : W2
  float* sh_m  = sh_w + 64 * WS;      // 64*64 : m2 = xp1 @ W2
  float* sh_ag = sh_m + 64 * 64;      // 64*64 : aggregation -> relu(h2)
  float* sh_sc = sh_ag + 64 * 64;     // 64
  int*   sh_ni = (int*)(sh_sc + 64);  // 128
  int*   sh_r2 = sh_ni + 128;         // 64

  const int g = blockIdx.x, tid = threadIdx.x;

  for (int i = tid; i < 64 * 64; i += 256) {
    int r = i >> 6, c = i & 63;
    sh_x[r * XS + c] = xp1[(size_t)g * 64 * 64 + i];
    sh_w[r * WS + c] = W2[i];
    sh_ag[i] = 0.f;
  }
  if (tid < 128) sh_ni[tid] = nid_g[g * 128 + tid];
  __syncthreads();

  // m2 = xp1 @ W2 : 16 output tiles of 16x16 over 8 waves, K=64 (fp32 WMMA)
  const int wave = tid >> 5, lane = tid & 31;
  const int half = lane >> 4, l16 = lane & 15;
  for (int t = wave; t < 16; t += 8) {
    int mt = t >> 2, nt = t & 3;
    v8f acc = {};
    for (int k = 0; k < 64; k += 4) {
      int kk = k + 2 * half;
      v2f a = { sh_x[(mt * 16 + l16) * XS + kk],
                sh_x[(mt * 16 + l16) * XS + kk + 1] };
      v2f b = { sh_w[kk * WS + nt * 16 + l16],
                sh_w[(kk + 1) * WS + nt * 16 + l16] };
      acc = WMMA_F32(a, b, acc);
    }
#pragma unroll
    for (int r = 0; r < 8; ++r)
      sh_m[(mt * 16 + 8 * half + r) * 64 + nt * 16 + l16] = acc[r];
  }
  __syncthreads();

  // masked edge aggregation on remapped ids (both endpoints must be kept)
  const long ebase = (long)g * EPG_;
  for (int e = tid; e < EPG_; e += 256) {
    int sl = esrc[ebase + e] - g * N0_;
    int dl = edst[ebase + e] - g * N0_;
    int ns = sh_ni[sl], nd = sh_ni[dl];
    if ((ns | nd) >= 0) {  // -1 has sign bit set -> OR < 0 if either dropped
      const float* srcv = sh_m + ns * 64;
      float* dstv = sh_ag + nd * 64;
      for (int dd = 0; dd < 64; ++dd) {
        int d = (dd + tid) & 63;
        atomicAdd(&dstv[d], srcv[d]);
      }
    }
  }
  __syncthreads();

  for (int i = tid; i < 64 * 64; i += 256) {
    float v = sh_ag[i] + b2[i & 63];
    sh_ag[i] = v > 0.f ? v : 0.f;
  }
  __syncthreads();

  if (tid < 64) {
    float dp = 0.f, pp = 0.f;
    for (int d = 0; d < 64; ++d) {
      float pv = p2[d];
      dp += sh_ag[tid * 64 + d] * pv;
      pp += pv * pv;
    }
    sh_sc[tid] = tanhf(dp * rsqrtf(pp));
  }
  __syncthreads();

  if (tid < 64) {
    float si = sh_sc[tid];
    int r = 0;
    for (int j = 0; j < 64; ++j) {
      float sj = sh_sc[j];
      r += (sj > si) || (sj == si && j < tid);
    }
    int id = (r < 32) ? r : -1;
    sh_r2[tid] = id;
    if (id >= 0) a2[(size_t)g * 32 + r] = 1.f / (1.f + expf(-si));
  }
  __syncthreads();

  // gate kept rows into sh_m (reuse), rank order
  for (int i = tid; i < 64 * 64; i += 256) {
    int node = i >> 6, d = i & 63;
    int r = sh_r2[node];
    if (r >= 0) sh_m[r * 64 + d] = sh_ag[i] * sh_sc[node];
  }
  __syncthreads();

  if (tid < 64) {
    float mx = -3.402823466e38f, smv = 0.f;
    for (int r = 0; r < 32; ++r) {
      float v = sh_m[r * 64 + tid];
      mx = v > mx ? v : mx;
      smv += v;
    }
    xg[(size_t)g * 128 + tid] = x1[(size_t)g * 128 + tid] + mx;
    xg[(size_t)g * 128 + 64 + tid] =
        x1[(size_t)g * 128 + 64 + tid] + smv * (1.f / 32.f);
  }
}

// ---------------------------------------------------------------------------
// Kernel 4: per-graph MLP head (fc1+bn1, fc2+bn2, fc3 + log_softmax).
// ---------------------------------------------------------------------------
__global__ void mlp_head(const float* __restrict__ xg,
                         const float* __restrict__ fc1w, const float* __restrict__ fc1b,
                         const float* __restrict__ g1, const float* __restrict__ be1,
                         const float* __restrict__ m1, const float* __restrict__ v1,
                         const float* __restrict__ fc2w, const float* __restrict__ fc2b,
                         const float* __restrict__ g2, const float* __restrict__ be2,
                         const float* __restrict__ m2, const float* __restrict__ v2,
                         const float* __restrict__ fc3w, const float* __restrict__ fc3b,
                         float* __restrict__ xy) {
  __shared__ float sx[128];
  __shared__ float sh1[256];
  __shared__ float sh2[512];
  __shared__ float red[2];
  const int g = blockIdx.x, tid = threadIdx.x;

  if (tid < 128) sx[tid] = xg[(size_t)g * 128 + tid];
  if (tid < 2) red[tid] = 0.f;
  __syncthreads();

  {  // fc1 (128 -> 256), relu, bn1(eval)
    float s = fc1b[tid];
    const float* w = fc1w + tid * 128;
    for (int i = 0; i < 128; ++i) s += w[i] * sx[i];
    s = s > 0.f ? s : 0.f;
    sh1[tid] = g1[tid] * (s - m1[tid]) * rsqrtf(v1[tid] + 1e-5f) + be1[tid];
  }
  __syncthreads();

  for (int o = tid; o < 512; o += 256) {  // fc2 (256 -> 512), relu, bn2
    float s = fc2b[o];
    const float* w = fc2w + o * 256;
    for (int i = 0; i < 256; ++i) s += w[i] * sh1[i];
    s = s > 0.f ? s : 0.f;
    sh2[o] = g2[o] * (s - m2[o]) * rsqrtf(v2[o] + 1e-5f) + be2[o];
  }
  __syncthreads();

  float p0 = 0.f, p1v = 0.f;  // fc3 (512 -> 2) via block reduction
  for (int i = tid; i < 512; i += 256) {
    p0 += fc3w[i] * sh2[i];
    p1v += fc3w[512 + i] * sh2[i];
  }
  atomicAdd(&red[0], p0);
  atomicAdd(&red[1], p1v);
  __syncthreads();

  if (tid == 0) {
    float l0 = red[0] + fc3b[0], l1 = red[1] + fc3b[1];
    float mx = l0 > l1 ? l0 : l1;
    float lse = mx + logf(expf(l0 - mx) + expf(l1 - mx));
    xy[(size_t)g * 2 + 0] = l0 - lse;
    xy[(size_t)g * 2 + 1] = l1 - lse;
  }
}

// ---------------------------------------------------------------------------
// Launch
// ---------------------------------------------------------------------------
extern "C" void kernel_launch(void* const* d_in, const int* in_sizes, int n_in,
                              void* d_out, int out_size, void* d_ws, size_t ws_size,
                              hipStream_t stream) {
  const float* x    = (const float*)d_in[0];
  const int*   ei   = (const int*)d_in[1];   // [2, E]
  // d_in[2] edge_attr, d_in[3] batch: unused by the reference math
  const float* W1   = (const float*)d_in[4]; // [128, 64]
  const float* b1   = (const float*)d_in[5];
  const float* p1   = (const float*)d_in[6];
  const float* W2   = (const float*)d_in[7]; // [64, 64]
  const float* b2   = (const float*)d_in[8];
  const float* p2   = (const float*)d_in[9];
  const float* fc1w = (const float*)d_in[10];
  const float* fc1b = (const float*)d_in[11];
  const float* bn1g = (const float*)d_in[12];
  const float* bn1b = (const float*)d_in[13];
  const float* bn1m = (const float*)d_in[14];
  const float* bn1v = (const float*)d_in[15];
  const float* fc2w = (const float*)d_in[16];
  const float* fc2b = (const float*)d_in[17];
  const float* bn2g = (const float*)d_in[18];
  const float* bn2b = (const float*)d_in[19];
  const float* bn2m = (const float*)d_in[20];
  const float* bn2v = (const float*)d_in[21];
  const float* fc3w = (const float*)d_in[22];
  const float* fc3b = (const float*)d_in[23];

  const int* esrc = ei;
  const int* edst = ei + E_;

  float* out = (float*)d_out;
  float* xy = out;                    // [2048, 2]
  float* a1 = out + B_ * 2;           // [2048, 64]
  float* a2 = a1 + (size_t)B_ * 64;   // [2048, 32]

  // Workspace carve-up (~99 MB), fully rewritten every call.
  float* h0  = (float*)d_ws;                        // [B*128, 64]
  float* xp1 = h0 + (size_t)B_ * N0_ * DIM_;        // [B*64, 64]
  float* x1  = xp1 + (size_t)B_ * 64 * DIM_;        // [B, 128]
  float* xg  = x1 + (size_t)B_ * 128;               // [B, 128]
  int*   nid = (int*)(xg + (size_t)B_ * 128);       // [B*128]

  // 1) h0 = x @ W1 (fp32 WMMA)
  {
    size_t smem = (size_t)(64 * (CIN_ + 2) + CIN_ * 65) * sizeof(float);
    gemm_wmma_n64<CIN_><<<(B_ * N0_) / 64, 128, smem, stream>>>(x, W1, h0);
  }
  // 2) per-graph conv1 aggregation + pool1 + readout1
  {
    size_t smem = (size_t)(2 * 128 * 64 + 128) * sizeof(float) + 128 * sizeof(int);
    graph_stage1<<<B_, 256, smem, stream>>>(h0, esrc, edst, b1, p1,
                                            xp1, nid, x1, a1);
  }
  // 3) per-graph conv2 (WMMA) + pool2 + readout2 + xg
  {
    size_t smem = (size_t)(64 * 66 + 64 * 65 + 64 * 64 + 64 * 64 + 64) * sizeof(float)
                + (size_t)(128 + 64) * sizeof(int);
    graph_stage2<<<B_, 256, smem, stream>>>(xp1, W2, esrc, edst, nid,
                                            b2, p2, x1, xg, a2);
  }
  // 4) MLP head + log_softmax
  mlp_head<<<B_, 256, 0, stream>>>(xg, fc1w, fc1b, bn1g, bn1b, bn1m, bn1v,
                                   fc2w, fc2b, bn2g, bn2b, bn2m, bn2v,
                                   fc3w, fc3b, xy);
}